// BaseModel_15049565405369
// MI455X (gfx1250) — compile-verified
//
#include <hip/hip_runtime.h>
#include <math.h>

// MI455X / gfx1250, wave32. All dense contractions via V_WMMA_F32_16X16X4_F32 (fp32 WMMA).
// CG table staged into LDS with GLOBAL_LOAD_ASYNC_TO_LDS_B128 (ASYNCcnt path).

typedef __attribute__((ext_vector_type(2))) float v2f;
typedef __attribute__((ext_vector_type(8))) float v8f;

__device__ __forceinline__ v8f wmma_f32_4(v2f a, v2f b, v8f c) {
  // (neg_a, A, neg_b, B, c_mod, C, reuse_a, reuse_b)
  return __builtin_amdgcn_wmma_f32_16x16x4_f32(false, a, false, b, (short)0, c, false, false);
}

// memory -> LDS async copy, 16B per lane; tracked by ASYNCcnt
__device__ __forceinline__ void async_copy_b128(unsigned lds_off, const float* base, unsigned gl_off) {
  asm volatile("global_load_async_to_lds_b128 %0, %1, %2"
               :: "v"(lds_off), "v"(gl_off), "s"(base) : "memory");
}
__device__ __forceinline__ void wait_async0() {
  asm volatile("s_wait_asynccnt 0" ::: "memory");
}
__device__ __forceinline__ unsigned lds_addr_of(const void* p) {
  // generic LDS pointers carry the LDS byte offset in the low 32 bits
  return (unsigned)(uintptr_t)p;
}

__device__ __forceinline__ unsigned hashu(unsigned x) {
  x ^= x >> 16; x *= 0x7feb352dU; x ^= x >> 15; x *= 0x846ca68bU; x ^= x >> 16; return x;
}
__device__ __forceinline__ int lof16(int i) { return (i == 0) ? 0 : ((i < 4) ? 1 : ((i < 9) ? 2 : 3)); }

// ---------------------------------------------------------------------------
// CGbig[m_all][i_all*16 + j_all]: dense 16x256 CG matrix; zero where no path.
// Deterministic pseudo-normal values (CG is module data, not an input).
// ---------------------------------------------------------------------------
__global__ void k_init_tables(float* __restrict__ cgbig) {
  for (int idx = threadIdx.x; idx < 4096; idx += blockDim.x) {
    int m  = idx >> 8;
    int ij = idx & 255;
    int i  = ij >> 4, j = ij & 15;
    int l1 = lof16(i), l2 = lof16(j), L = lof16(m);
    int lo = l1 - l2; if (lo < 0) lo = -lo;
    float v = 0.0f;
    if (L >= lo && L <= l1 + l2) {
      unsigned h1 = hashu((unsigned)idx * 2654435761u + 12345u);
      unsigned h2 = hashu((unsigned)idx * 2246822519u + 98765u);
      float u1 = ((float)h1 + 1.0f) * 2.3283064365386963e-10f;
      float u2 = (float)h2 * 2.3283064365386963e-10f;
      v = 0.2f * sqrtf(-2.0f * logf(u1)) * cosf(6.28318530718f * u2);
    }
    cgbig[idx] = v;
  }
}

// ---------------------------------------------------------------------------
// Edge geometry + radial basis + SH + first message pass (atomic scatter).
// One wave per edge, 8 waves / 256-thread block.
// ---------------------------------------------------------------------------
__global__ void __launch_bounds__(256) k_edge_mp1(
    const float* __restrict__ pos, const float* __restrict__ emb,
    const float* __restrict__ Wrad, const int* __restrict__ species,
    const int* __restrict__ snd, const int* __restrict__ rcv,
    float* __restrict__ Rl, float* __restrict__ Sh, float* __restrict__ feats)
{
  int w = threadIdx.x >> 5, lane = threadIdx.x & 31;
  int e = blockIdx.x * 8 + w;
  int s = snd[e], r = rcv[e];
  float rx = pos[r * 3 + 0] - pos[s * 3 + 0];
  float ry = pos[r * 3 + 1] - pos[s * 3 + 1];
  float rz = pos[r * 3 + 2] - pos[s * 3 + 2];
  float d  = sqrtf(rx * rx + ry * ry + rz * rz + 1e-8f);
  float inv = 1.0f / d;
  float x = rx * inv, y = ry * inv, z = rz * inv;
  float fc = (d < 5.0f) ? 0.5f * (cosf(0.6283185307f * d) + 1.0f) : 0.0f;

  float R[4];
#pragma unroll
  for (int n = 0; n < 4; ++n) {
    float c = (5.0f / 3.0f) * (float)n;
    float t = d - c;
    R[n] = expf(-0.5f * t * t) * fc;
  }

  // full Rl (constant indices only -> stays in registers), stored by lane 0
  float rl[16];
#pragma unroll
  for (int l = 0; l < 4; ++l)
#pragma unroll
    for (int j = 0; j < 4; ++j) {
      float a = 0.0f;
#pragma unroll
      for (int n = 0; n < 4; ++n) a += R[n] * Wrad[l * 16 + n * 4 + j];
      rl[l * 4 + j] = a;
    }

  float xx = x * x, yy = y * y, zz = z * z;
  float sh[16];
  sh[0] = 1.0f; sh[1] = x; sh[2] = y; sh[3] = z;
  sh[4] = x * y; sh[5] = y * z; sh[6] = 3.0f * zz - 1.0f; sh[7] = x * z; sh[8] = xx - yy;
  sh[9]  = y * (3.0f * xx - yy); sh[10] = x * y * z; sh[11] = y * (5.0f * zz - 1.0f);
  sh[12] = z * (5.0f * zz - 3.0f); sh[13] = x * (5.0f * zz - 1.0f);
  sh[14] = z * (xx - yy); sh[15] = x * (xx - 3.0f * yy);
#pragma unroll
  for (int m = 0; m < 16; ++m) sh[m] *= 0.5f;  // NU_SCALING folded in

  if (lane == 0) {
#pragma unroll
    for (int m = 0; m < 16; ++m) {
      Sh[(size_t)e * 16 + m] = sh[m];
      Rl[(size_t)e * 16 + m] = rl[m];
    }
  }

  // per-thread radial values for this lane's two channels k=lane and k=lane+32
  int ja = lane >> 4;            // n index for k=lane   (0 or 1)
  int jb = (lane + 32) >> 4;     // n index for k=lane+32 (2 or 3)
  float rla[4], rlb[4];
#pragma unroll
  for (int l = 0; l < 4; ++l) {
    float aa = 0.0f, bb = 0.0f;
#pragma unroll
    for (int n = 0; n < 4; ++n) {
      aa += R[n] * Wrad[l * 16 + n * 4 + ja];
      bb += R[n] * Wrad[l * 16 + n * 4 + jb];
    }
    rla[l] = aa; rlb[l] = bb;
  }

  int sp = species[r];
  float em = emb[sp * 16 + (lane & 15)];  // (lane+32)&15 == lane&15
  const int lofm[16] = {0, 1, 1, 1, 2, 2, 2, 2, 2, 3, 3, 3, 3, 3, 3, 3};
  float* fb = feats + (size_t)s * 1024;
#pragma unroll
  for (int m = 0; m < 16; ++m) {
    int l = lofm[m];
    float bse = 0.1f * sh[m] * em;  // MP_SCALING folded in
    atomicAdd(&fb[m * 64 + lane],      bse * rla[l]);
    atomicAdd(&fb[m * 64 + lane + 32], bse * rlb[l]);
  }
}

// ---------------------------------------------------------------------------
// cg_iterate x2 + embed, one atom per wave. new = CGbig(16x256) @ P(256x64),
// P[ij,k] = f[i,k]*base[j,k] built on the fly from LDS. WMMA f32 16x16x4.
// mode=1: write full emb_feats (16x64); mode=0: write only m=0 row (head input)
// ---------------------------------------------------------------------------
__global__ void __launch_bounds__(128) k_cg_embed(
    const float* __restrict__ fin, const float* __restrict__ cg,
    const float* __restrict__ emb, const int* __restrict__ species,
    float* __restrict__ fout, int mode)
{
  __shared__ __align__(16) float cgl[4096];  // 16 KB, block-shared
  __shared__ float fbuf[4][1024];            // 16 KB, per-wave atom features
  __shared__ float bbuf[4][1024];            // 16 KB, per-wave base features
  int w = threadIdx.x >> 5, lane = threadIdx.x & 31;
  int atom = blockIdx.x * 4 + w;

  // async memory->LDS copy of the CG table (no VGPR round-trip)
  {
    unsigned lbase = lds_addr_of(cgl);
#pragma unroll
    for (int c = 0; c < 8; ++c) {
      unsigned off = (unsigned)(threadIdx.x + c * 128) * 16u;
      async_copy_b128(lbase + off, cg, off);
    }
  }
  for (int off = lane; off < 1024; off += 32) {
    float v = fin[(size_t)atom * 1024 + off];
    fbuf[w][off] = v;
    bbuf[w][off] = v;
  }
  wait_async0();
  __syncthreads();

  int kl = lane & 15;
  int hi = (lane < 16) ? 0 : 2;
  int m0 = (lane < 16) ? 0 : 8;

  for (int it = 0; it < 2; ++it) {
    v8f acc[4] = {};
#pragma unroll 4
    for (int sx = 0; sx < 64; ++sx) {      // K = 256, 4 per WMMA
      int ij0 = 4 * sx + hi;
      v2f a;
      a.x = cgl[kl * 256 + ij0];
      a.y = cgl[kl * 256 + ij0 + 1];
#pragma unroll
      for (int t = 0; t < 4; ++t) {        // 4 N-tiles over 64 channels
        int k = t * 16 + kl;
        v2f b;
        b.x = fbuf[w][((ij0)     >> 4) * 64 + k] * bbuf[w][((ij0)     & 15) * 64 + k];
        b.y = fbuf[w][((ij0 + 1) >> 4) * 64 + k] * bbuf[w][((ij0 + 1) & 15) * 64 + k];
        acc[t] = wmma_f32_4(a, b, acc[t]);
      }
    }
    __syncthreads();
#pragma unroll
    for (int t = 0; t < 4; ++t)
#pragma unroll
      for (int rr = 0; rr < 8; ++rr)
        fbuf[w][(m0 + rr) * 64 + t * 16 + kl] += 0.5f * acc[t][rr];
    __syncthreads();
  }

  int sp = species[atom];
  if (mode) {
    for (int off = lane; off < 1024; off += 32)
      fout[(size_t)atom * 1024 + off] = fbuf[w][off] * emb[sp * 16 + (off & 15)];
  } else {
    for (int k = lane; k < 64; k += 32)
      fout[(size_t)atom * 64 + k] = fbuf[w][k] * emb[sp * 16 + (k & 15)];
  }
}

// ---------------------------------------------------------------------------
// Second message pass: per-edge U^{l1}(16x16) @ G(16x64), Rl folded into B
// columns, single WMMA accumulator over l1; atomic scatter to senders.
// One wave per edge, 8 waves / block.
// ---------------------------------------------------------------------------
__global__ void __launch_bounds__(256) k_mp2(
    const float* __restrict__ embA, const float* __restrict__ cg,
    const float* __restrict__ Rl, const float* __restrict__ Sh,
    const int* __restrict__ snd, const int* __restrict__ rcv,
    float* __restrict__ mp)
{
  __shared__ __align__(16) float cgl[4096];  // 16 KB
  __shared__ float shl[8][16];
  __shared__ float rll[8][16];
  __shared__ float ul[8][1024];              // U[l1][m][j], 32 KB
  int w = threadIdx.x >> 5, lane = threadIdx.x & 31;
  int e = blockIdx.x * 8 + w;

  // async memory->LDS copy of the CG table
  {
    unsigned lbase = lds_addr_of(cgl);
#pragma unroll
    for (int c = 0; c < 4; ++c) {
      unsigned off = (unsigned)(threadIdx.x + c * 256) * 16u;
      async_copy_b128(lbase + off, cg, off);
    }
  }
  int s = snd[e], r = rcv[e];
  const float* g = embA + (size_t)r * 1024;
  __builtin_prefetch(g, 0, 0);               // global_prefetch: hide gather latency
  __builtin_prefetch(g + 512, 0, 0);

  if (lane < 16) shl[w][lane] = Sh[(size_t)e * 16 + lane];
  else           rll[w][lane - 16] = Rl[(size_t)e * 16 + (lane - 16)];
  wait_async0();
  __syncthreads();

  // U^{l1}[m][j] = 0.1 * sum_i sh[i]*CG[m][i*16+j], i in l1 block
#pragma unroll
  for (int l1 = 0; l1 < 4; ++l1) {
    const int i0 = (l1 == 0) ? 0 : ((l1 == 1) ? 1 : ((l1 == 2) ? 4 : 9));
    const int ni = 2 * l1 + 1;
    for (int cc = lane; cc < 256; cc += 32) {
      int m = cc >> 4, j = cc & 15;
      float u = 0.0f;
#pragma unroll
      for (int i = 0; i < ni; ++i)
        u += shl[w][i0 + i] * cgl[m * 256 + (i0 + i) * 16 + j];
      ul[w][l1 * 256 + cc] = 0.1f * u;
    }
  }
  __syncthreads();

  float rlr[16];
#pragma unroll
  for (int i = 0; i < 16; ++i) rlr[i] = rll[w][i];

  int kl = lane & 15;
  int hi = (lane < 16) ? 0 : 2;
  v8f acc[4] = {};
#pragma unroll
  for (int sx = 0; sx < 4; ++sx) {         // K = 16 (j_all)
    int j0 = 4 * sx + hi;
    v2f a[4];
#pragma unroll
    for (int l1 = 0; l1 < 4; ++l1) {
      a[l1].x = ul[w][l1 * 256 + kl * 16 + j0];
      a[l1].y = ul[w][l1 * 256 + kl * 16 + j0 + 1];
    }
#pragma unroll
    for (int t = 0; t < 4; ++t) {          // channel tiles; radial n == t
      int k = t * 16 + kl;
      float g0 = g[(size_t)j0 * 64 + k];
      float g1 = g[(size_t)(j0 + 1) * 64 + k];
#pragma unroll
      for (int l1 = 0; l1 < 4; ++l1) {
        float rs = rlr[l1 * 4 + t];
        v2f b; b.x = g0 * rs; b.y = g1 * rs;
        acc[t] = wmma_f32_4(a[l1], b, acc[t]);
      }
    }
  }

  int m0 = (lane < 16) ? 0 : 8;
  float* mb = mp + (size_t)s * 1024;
#pragma unroll
  for (int t = 0; t < 4; ++t)
#pragma unroll
    for (int rr = 0; rr < 8; ++rr)
      atomicAdd(&mb[(m0 + rr) * 64 + t * 16 + kl], acc[t][rr]);
}

// ---------------------------------------------------------------------------
// Head: h = silu(f0 @ W_head + b_head); energy = h @ W_last + b_last.
// One 16-atom row block per 128-thread block; wave w owns channel tile w.
// Cross-wave reduction via LDS float atomics; single biased store.
// ---------------------------------------------------------------------------
__global__ void __launch_bounds__(128) k_head(
    const float* __restrict__ f0, const float* __restrict__ Wh,
    const float* __restrict__ bh, const float* __restrict__ Wl,
    const float* __restrict__ blast, float* __restrict__ out)
{
  __shared__ float osum[16];
  int w = threadIdx.x >> 5, lane = threadIdx.x & 31;
  int row0 = blockIdx.x * 16;
  int kl = lane & 15;
  int hi = (lane < 16) ? 0 : 2;
  int n = w * 16 + kl;

  if (threadIdx.x < 16) osum[threadIdx.x] = 0.0f;
  __syncthreads();

  v8f acc = {};
#pragma unroll
  for (int s2 = 0; s2 < 16; ++s2) {        // K = 64
    int kk = 4 * s2 + hi;
    v2f a, b;
    a.x = f0[(size_t)(row0 + kl) * 64 + kk];
    a.y = f0[(size_t)(row0 + kl) * 64 + kk + 1];
    b.x = Wh[(size_t)kk * 64 + n];
    b.y = Wh[(size_t)(kk + 1) * 64 + n];
    acc = wmma_f32_4(a, b, acc);
  }

  float bias = bh[n];
  float wl = Wl[n];
#pragma unroll
  for (int rr = 0; rr < 8; ++rr) {
    float zv = acc[rr] + bias;
    float hv = zv / (1.0f + expf(-zv));    // silu
    float v = hv * wl;
    v += __shfl_xor(v, 1, 16);
    v += __shfl_xor(v, 2, 16);
    v += __shfl_xor(v, 4, 16);
    v += __shfl_xor(v, 8, 16);
    if (kl == 0) atomicAdd(&osum[(lane < 16) ? rr : rr + 8], v);  // ds_add_f32
  }
  __syncthreads();
  if (threadIdx.x < 16) out[row0 + threadIdx.x] = osum[threadIdx.x] + blast[0];
}

// ---------------------------------------------------------------------------
extern "C" void kernel_launch(void* const* d_in, const int* in_sizes, int n_in,
                              void* d_out, int out_size, void* d_ws, size_t ws_size,
                              hipStream_t stream)
{
  const float* pos   = (const float*)d_in[0];
  const float* emb   = (const float*)d_in[1];
  const float* Wrad  = (const float*)d_in[2];
  const float* Whead = (const float*)d_in[3];
  const float* bhead = (const float*)d_in[4];
  const float* Wlast = (const float*)d_in[5];
  const float* blast = (const float*)d_in[6];
  const int* species = (const int*)d_in[7];
  const int* snd     = (const int*)d_in[8];
  const int* rcv     = (const int*)d_in[9];
  int nAtoms = in_sizes[7];
  int nEdges = in_sizes[8];

  char* ws = (char*)d_ws;
  size_t off = 0;
  float* cgbig  = (float*)(ws + off); off += 65536;                               // 16KB used
  float* Rl     = (float*)(ws + off); off += (size_t)nEdges * 16 * sizeof(float);
  float* Sh     = (float*)(ws + off); off += (size_t)nEdges * 16 * sizeof(float);
  float* featsA = (float*)(ws + off); off += (size_t)nAtoms * 1024 * sizeof(float);
  float* embA   = (float*)(ws + off); off += (size_t)nAtoms * 1024 * sizeof(float);
  float* mpB    = (float*)(ws + off); off += (size_t)nAtoms * 1024 * sizeof(float);
  float* f0     = (float*)(ws + off); off += (size_t)nAtoms * 64 * sizeof(float);

  k_init_tables<<<1, 256, 0, stream>>>(cgbig);
  hipMemsetAsync(featsA, 0, (size_t)nAtoms * 1024 * sizeof(float), stream);
  hipMemsetAsync(mpB,    0, (size_t)nAtoms * 1024 * sizeof(float), stream);

  k_edge_mp1<<<nEdges / 8, 256, 0, stream>>>(pos, emb, Wrad, species, snd, rcv,
                                             Rl, Sh, featsA);
  k_cg_embed<<<nAtoms / 4, 128, 0, stream>>>(featsA, cgbig, emb, species, embA, 1);
  k_mp2<<<nEdges / 8, 256, 0, stream>>>(embA, cgbig, Rl, Sh, snd, rcv, mpB);
  k_cg_embed<<<nAtoms / 4, 128, 0, stream>>>(mpB, cgbig, emb, species, f0, 0);
  k_head<<<nAtoms / 16, 128, 0, stream>>>(f0, Whead, bhead, Wlast, blast, (float*)d_out);
}